// CropSampler_8512625181123
// MI455X (gfx1250) — compile-verified
//
#include <hip/hip_runtime.h>

#define CROP 256
#define NB 16
#define TILE_X 32  // q / width-direction (fast output dim)
#define TILE_Y 16  // p / height-direction
// Max staged window per channel: x-span 31*(700/255)=85.1 -> <=88 cols; y-span 15*step -> <=44 rows.
#define MAXCX 90
#define MAXCY 46

typedef unsigned int v4u __attribute__((ext_vector_type(4)));
typedef int          v8i __attribute__((ext_vector_type(8)));
typedef int          v4i __attribute__((ext_vector_type(4)));

__device__ __forceinline__ int rfl(int v) { return __builtin_amdgcn_readfirstlane(v); }

// Matches reference: (arange(cs)/255) * bbox_size + t, separate mul/add (no contraction)
__device__ __forceinline__ float coordf(int i, float size, float t) {
#pragma clang fp contract(off)
  return ((float)i / 255.0f) * size + t;
}

__global__ void __launch_bounds__(256) crop_sample_kernel(
    const float* __restrict__ img, const float* __restrict__ center,
    const float* __restrict__ bbox, const int* __restrict__ heights,
    const int* __restrict__ widths, const int* __restrict__ starts,
    float* __restrict__ out_images, float* __restrict__ out_flat) {
  __shared__ float tile[3 * MAXCX * MAXCY];  // 49,680 bytes

  const int b = blockIdx.z;
  const float size = bbox[b];
  const float txo = center[2 * b + 0] - size * 0.5f;
  const float tyo = center[2 * b + 1] - size * 0.5f;
  const int w = widths[b];
  const int h = heights[b];
  const int start = starts[b];
  const int wmax = w - 1, hmax = h - 1;
  const int hw = h * w;

  const int q0 = blockIdx.x * TILE_X;
  const int p0 = blockIdx.y * TILE_Y;

  // Uniform staged-window bounds (clamped to image); covers every clipped corner read.
  const float x_first = coordf(q0, size, txo);
  const float x_last  = coordf(q0 + TILE_X - 1, size, txo);
  const float y_first = coordf(p0, size, tyo);
  const float y_last  = coordf(p0 + TILE_Y - 1, size, tyo);
  const int gx0 = min(max((int)floorf(x_first), 0), wmax);
  const int gx1 = min(max((int)floorf(x_last) + 1, 0), wmax);
  const int gy0 = min(max((int)floorf(y_first), 0), hmax);
  const int gy1 = min(max((int)floorf(y_last) + 1, 0), hmax);
  const int cols = min(gx1 - gx0 + 1, MAXCX);
  const int rows = min(gy1 - gy0 + 1, MAXCY);
  const int cstr = rows * cols;  // LDS channel stride (elements)

  // ---- Stage 1: wave 0 issues 3 TDM 2D-tile loads (one per channel). No wait yet:
  // the DMA runs while all waves do the coordinate/weight VALU below.
  if ((threadIdx.x >> 5) == 0) {
    const unsigned lds_base =
        (unsigned)(unsigned long long)(const void*)&tile[0];  // low 32 bits = LDS byte offset
    const int scols = rfl(cols);
    const int srows = rfl(rows);
    const int sw = rfl(w);
    const int scstr = scols * srows;
    for (int ch = 0; ch < 3; ++ch) {
      const unsigned long long ga = (unsigned long long)(const void*)(
          img + (long long)start + (long long)ch * hw + (long long)gy0 * w + gx0);
      const unsigned ga_lo = (unsigned)rfl((int)(ga & 0xFFFFFFFFull));
      const unsigned ga_hi = (unsigned)rfl((int)(ga >> 32));
      const unsigned lds_addr =
          (unsigned)rfl((int)(lds_base + (unsigned)ch * (unsigned)scstr * 4u));

      v4u g0;
      g0.x = 1u;        // D# group0: count=1, user descriptor
      g0.y = lds_addr;  // lds_addr[31:0]
      g0.z = ga_lo;     // global_addr[31:0]
      g0.w = (ga_hi & 0x01FFFFFFu) | 0x80000000u;  // global_addr[56:32] | type=2

      v8i g1;
      g1.s0 = 0x00020000;              // workgroup_mask=0, data_size=2 (4B), no pad/iterate
      g1.s1 = (scols & 0xFFFF) << 16;  // tensor_dim0[15:0] (bits 79:48)
      g1.s2 = (srows & 0xFFFF) << 16;  // tensor_dim1[15:0] (bits 111:80)
      g1.s3 = (scols & 0xFFFF) << 16;  // tile_dim0 (bits 127:112)
      g1.s4 = (srows & 0xFFFF);        // tile_dim1 (bits 143:128), tile_dim2=0 (2D)
      g1.s5 = sw;                      // tensor_dim0_stride[31:0] = image width
      g1.s6 = 0;                       // stride0 hi / stride1 lo
      g1.s7 = 0;

      v4i gz = {0, 0, 0, 0};  // groups 2/3 unused for 2D tile (tile_dim3/4 = 0)
#if __clang_major__ >= 23
      v8i gz8 = {0, 0, 0, 0, 0, 0, 0, 0};
      __builtin_amdgcn_tensor_load_to_lds(g0, g1, gz, gz, gz8, 0);
#else
      __builtin_amdgcn_tensor_load_to_lds(g0, g1, gz, gz, 0);
#endif
    }
  }

  // ---- Stage 2 (overlapped with DMA): coordinates, weights, masks, and the `flat` output.
  const int tq = threadIdx.x & (TILE_X - 1);
  const int tp = threadIdx.x >> 5;  // 0..7; each thread covers p rows {tp, tp+8}
  const int q = q0 + tq;

  const float x = coordf(q, size, txo);
  const int x0 = (int)floorf(x);
  // Reference bug reproduced: x1/y1 in-bounds checks reuse x0/y0 -> one shared mask.
  const bool inx = (x0 >= 0) && (x0 <= wmax);
  const int x0c = min(max(x0, 0), wmax);
  const int x1c = min(max(x0 + 1, 0), wmax);
  const float wx1 = (float)x1c - x;
  const float wx0 = x - (float)x0c;
  const int lx0 = x0c - gx0, lx1 = x1c - gx0;

  int   jj[2], lly0[2], lly1[2];
  float wwy1[2], wwy0[2], msk[2];
#pragma unroll
  for (int k = 0; k < 2; ++k) {
    const int p = p0 + tp + 8 * k;
    const float y = coordf(p, size, tyo);
    const int y0 = (int)floorf(y);
    const bool iny = (y0 >= 0) && (y0 <= hmax);
    msk[k] = (inx && iny) ? 1.0f : 0.0f;
    const int y0c = min(max(y0, 0), hmax);
    const int y1c = min(max(y0 + 1, 0), hmax);
    wwy1[k] = (float)y1c - y;
    wwy0[k] = y - (float)y0c;
    lly0[k] = y0c - gy0;
    lly1[k] = y1c - gy0;
    jj[k] = p * CROP + q;
    float2 xy;
    xy.x = x;
    xy.y = y;
    *(float2*)(out_flat + 2 * ((size_t)b * CROP * CROP + jj[k])) = xy;
  }

  // ---- Stage 3: issuing wave drains TENSORcnt, then workgroup barrier publishes the tile.
  if ((threadIdx.x >> 5) == 0) {
    __builtin_amdgcn_s_wait_tensorcnt(0);
  }
  __syncthreads();

  // ---- Stage 4: bilinear sample from LDS, coalesced image stores.
  float* ob = out_images + (size_t)b * 3 * CROP * CROP;
#pragma unroll
  for (int k = 0; k < 2; ++k) {
    const int r0 = lly0[k] * cols, r1 = lly1[k] * cols;
    const float m = msk[k];
#pragma unroll
    for (int ch = 0; ch < 3; ++ch) {
      const float* t = tile + ch * cstr;
      const float Ia = t[r0 + lx0] * m;
      const float Ib = t[r0 + lx1] * m;
      const float Ic = t[r1 + lx0] * m;
      const float Id = t[r1 + lx1] * m;
      const float f1 = wx1 * Ia + wx0 * Ib;
      const float f2 = wx1 * Ic + wx0 * Id;
      ob[ch * CROP * CROP + jj[k]] = wwy1[k] * f1 + wwy0[k] * f2;
    }
  }
}

__global__ void crop_transforms_kernel(const float* __restrict__ center,
                                       const float* __restrict__ bbox,
                                       float* __restrict__ out_t,
                                       float* __restrict__ out_h) {
  const int b = threadIdx.x;
  if (b < NB) {
    const float size = bbox[b];
    const float txo = center[2 * b + 0] - size * 0.5f;
    const float tyo = center[2 * b + 1] - size * 0.5f;
    float* T = out_t + b * 9;
    T[0] = size; T[1] = 0.0f; T[2] = txo;
    T[3] = 0.0f; T[4] = size; T[5] = tyo;
    T[6] = 0.0f; T[7] = 0.0f; T[8] = 1.0f;
    const float scale = 2.0f * (float)(CROP - 1) / size;
    float* H = out_h + b * 9;
    H[0] = scale; H[1] = 0.0f; H[2] = -txo * scale - 1.0f;
    H[3] = 0.0f;  H[4] = scale; H[5] = -tyo * scale - 1.0f;
    H[6] = 0.0f;  H[7] = 0.0f;  H[8] = 1.0f;
  }
}

extern "C" void kernel_launch(void* const* d_in, const int* in_sizes, int n_in,
                              void* d_out, int out_size, void* d_ws, size_t ws_size,
                              hipStream_t stream) {
  const float* img    = (const float*)d_in[0];
  const float* center = (const float*)d_in[1];
  const float* bbox   = (const float*)d_in[2];
  const int* heights  = (const int*)d_in[3];
  const int* widths   = (const int*)d_in[4];
  const int* starts   = (const int*)d_in[5];

  float* out        = (float*)d_out;
  float* out_images = out;                                       // 16*3*256*256
  float* out_flat   = out_images + (size_t)NB * 3 * CROP * CROP; // 16*65536*2
  float* out_t      = out_flat + (size_t)NB * CROP * CROP * 2;   // 16*9
  float* out_h      = out_t + NB * 9;                            // 16*9

  dim3 grid(CROP / TILE_X, CROP / TILE_Y, NB);  // 8 x 16 x 16 blocks
  crop_sample_kernel<<<grid, 256, 0, stream>>>(img, center, bbox, heights, widths,
                                               starts, out_images, out_flat);
  crop_transforms_kernel<<<1, 32, 0, stream>>>(center, bbox, out_t, out_h);
}